// skip_gram_23072564314584
// MI455X (gfx1250) — compile-verified
//
#include <hip/hip_runtime.h>

typedef __attribute__((ext_vector_type(2))) float v2f;
typedef __attribute__((ext_vector_type(4))) float v4f;
typedef __attribute__((ext_vector_type(8))) float v8f;

#define DIM 128
#define LMAX 20
#define ROWS_PER_WAVE 16
#define WAVES_PER_BLOCK 8
#define ROWS_PER_BLOCK (ROWS_PER_WAVE * WAVES_PER_BLOCK)   // 128

// Stable log(sigmoid(x)) = min(x,0) - log1p(exp(-|x|))
__device__ __forceinline__ float log_sigmoid(float x) {
    return fminf(x, 0.0f) - log1pf(__expf(-fabsf(x)));
}

__global__ __launch_bounds__(256) void hs_skipgram_loss_kernel(
    const float* __restrict__ W0,        // [N_NODES, 128] word embeddings
    const float* __restrict__ W1,        // [N_NODES, 128] node embeddings
    const int*   __restrict__ target,    // [B]
    const int*   __restrict__ context,   // [B]
    const int*   __restrict__ codes,     // [B, LMAX]
    const int*   __restrict__ lengths,   // [B]
    float*       __restrict__ partials,  // [gridDim.x] block partial sums
    int B)
{
    const int lane = threadIdx.x & 31;
    const int wave = threadIdx.x >> 5;
    const int half = lane >> 4;          // 0: lanes 0-15, 1: lanes 16-31
    const int sub  = lane & 15;          // tile row (A) / tile col (B) id

    const int tileBase = blockIdx.x * ROWS_PER_BLOCK + wave * ROWS_PER_WAVE;
    // Branch-free clamp keeps EXEC all-1s for WMMA even on a ragged tail.
    const int rowIdx = min(tileBase + sub, B - 1);

    const long long t = (long long)target[rowIdx];
    const long long c = (long long)context[rowIdx];
    const float* __restrict__ rowT = W0 + t * DIM;   // A-matrix source (syn0)
    const float* __restrict__ rowC = W1 + c * DIM;   // B-matrix source (syn1)

    // Accumulate syn0_tile (16x128) x syn1_tile^T (128x16) via 32 chained
    // V_WMMA_F32_16X16X4_F32; we only need the diagonal (per-row dots).
    v8f acc = {};
    #pragma unroll
    for (int kk = 0; kk < DIM; kk += 4) {
        // A 16x4 f32 layout: VGPR0 = {K=kk+0 | K=kk+2}, VGPR1 = {K=kk+1 | K=kk+3}
        // lanes 0-15 take K offsets {0,1}; lanes 16-31 take {2,3} -> contiguous f32x2
        v2f a = *(const v2f*)(rowT + kk + (half << 1));

        // B 4x16 f32 layout: VGPR0 = {K=kk+0 | K=kk+1}, VGPR1 = {K=kk+2 | K=kk+3}
        // lane needs K = kk+half and kk+2+half from its column row -> select from f32x4
        v4f b4 = *(const v4f*)(rowC + kk);
        v2f b;
        b.x = half ? b4.y : b4.x;
        b.y = half ? b4.w : b4.z;

        acc = __builtin_amdgcn_wmma_f32_16x16x4_f32(
            /*neg_a=*/false, a, /*neg_b=*/false, b,
            /*c_mod=*/(short)0, acc, /*reuse_a=*/false, /*reuse_b=*/false);
    }

    // Diagonal extraction (C/D 16x16 f32 layout):
    //   m in 0..7  -> lane m,      VGPR m
    //   m in 8..15 -> lane m+16,   VGPR m-8
    const bool hasDiag = (lane < 8) | (lane >= 24);
    const int  m       = (lane < 8) ? lane : (lane - 16);   // tile row 0..15
    const int  ci      = (lane < 8) ? lane : (lane - 24);   // acc element
    float dot = 0.0f;
    #pragma unroll
    for (int i = 0; i < 8; ++i) dot = (ci == i) ? acc[i] : dot;

    // Per-row Huffman-path loss: -sum_l valid(l) * log_sigmoid(sign_l * dot)
    float loss = 0.0f;
    const int gRow = tileBase + m;
    if (hasDiag && gRow < B) {
        const int len = lengths[gRow];
        const int* __restrict__ cr = codes + (long long)gRow * LMAX;
        #pragma unroll
        for (int l = 0; l < LMAX; ++l) {
            if (l < len) {
                const float sign = 1.0f - 2.0f * (float)cr[l];
                loss -= log_sigmoid(sign * dot);
            }
        }
    }

    // Wave32 butterfly reduction, then 8-wave LDS reduction -> one partial/block
    #pragma unroll
    for (int off = 16; off > 0; off >>= 1)
        loss += __shfl_xor(loss, off, 32);

    __shared__ float sacc[WAVES_PER_BLOCK];
    if (lane == 0) sacc[wave] = loss;
    __syncthreads();
    if (threadIdx.x == 0) {
        float s = 0.0f;
        #pragma unroll
        for (int i = 0; i < WAVES_PER_BLOCK; ++i) s += sacc[i];
        partials[blockIdx.x] = s;
    }
}

// Deterministic final reduction of block partials (no float atomics).
__global__ __launch_bounds__(128) void hs_reduce_kernel(
    const float* __restrict__ partials, int n, float* __restrict__ out)
{
    float s = 0.0f;
    for (int i = threadIdx.x; i < n; i += blockDim.x) s += partials[i];
    #pragma unroll
    for (int off = 16; off > 0; off >>= 1)
        s += __shfl_xor(s, off, 32);

    __shared__ float sacc[4];
    const int lane = threadIdx.x & 31;
    const int wave = threadIdx.x >> 5;
    if (lane == 0) sacc[wave] = s;
    __syncthreads();
    if (threadIdx.x == 0) {
        float t = 0.0f;
        #pragma unroll
        for (int i = 0; i < 4; ++i) t += sacc[i];
        out[0] = t;
    }
}

extern "C" void kernel_launch(void* const* d_in, const int* in_sizes, int n_in,
                              void* d_out, int out_size, void* d_ws, size_t ws_size,
                              hipStream_t stream) {
    const float* W0      = (const float*)d_in[0];
    const float* W1      = (const float*)d_in[1];
    const int*   target  = (const int*)d_in[2];
    const int*   context = (const int*)d_in[3];
    const int*   codes   = (const int*)d_in[4];
    const int*   lengths = (const int*)d_in[5];
    float*       out      = (float*)d_out;
    float*       partials = (float*)d_ws;

    const int B = in_sizes[2];                       // 16384
    const int nblocks = (B + ROWS_PER_BLOCK - 1) / ROWS_PER_BLOCK;  // 128

    hs_skipgram_loss_kernel<<<nblocks, 256, 0, stream>>>(
        W0, W1, target, context, codes, lengths, partials, B);
    hs_reduce_kernel<<<1, 128, 0, stream>>>(partials, nblocks, out);
}